// EncoderLayer_1460288881504
// MI455X (gfx1250) — compile-verified
//
#include <hip/hip_runtime.h>
#include <hip/hip_bf16.h>
#include <cstddef>

#define LSEQ   2048
#define DMODEL 256
#define DINNER 256
#define DSTATE 256
#define DTRANK 16
#define DXP    (DTRANK + 2*DSTATE)   /* 528 */
#define DFF    1024
#define NROWS  (2*LSEQ)              /* B*L = 4096 */

typedef __attribute__((ext_vector_type(2))) float v2f;
typedef __attribute__((ext_vector_type(8))) float v8f;

__device__ __forceinline__ float silu_f(float v)     { return v / (1.f + __expf(-v)); }
__device__ __forceinline__ float softplus_f(float v) { return (v > 20.f) ? v : log1pf(__expf(v)); }

// ---------------------------------------------------------------------------
// Y[M, Ncols] = X[M,K] @ W[Ncols,K]^T  using V_WMMA_F32_16X16X4_F32.
// One wave computes a 16(M) x (16*NT)(N) tile: NT independent accumulators.
// Ncols must be a multiple of 16*NT; K must be a multiple of 8 (true here:
// K in {16, 256, 1024}).  Inner loop is branch-free (EXEC all-1s around every
// v_wmma) and software-pipelined with prefetch distance 2 using two
// statically-named buffer sets; unroll-2 of the main loop removes the
// buffer-rotation movs.  32-bit offsets from uniform bases give the
// SADDR + voffset global-load form.
// EPI: 0=none, 1=+bias, 2=+bias,relu, 3=+bias,softplus.
// FLIP: read X rows time-reversed within each batch of LSEQ rows.
// ---------------------------------------------------------------------------
template<int EPI, bool FLIP, int NT>
__global__ __launch_bounds__(256) void gemm_wmma_f32(
    const float* __restrict__ X, int lda,
    const float* __restrict__ W,
    const float* __restrict__ bias,
    float* __restrict__ Y, int ldy,
    int M, int Ncols, int K)
{
    const int wave = blockIdx.x * 8 + (threadIdx.x >> 5);
    const int lane = threadIdx.x & 31;
    const int nG   = Ncols / (16 * NT);
    const int tm   = wave / nG;
    if (tm >= (M >> 4)) return;                 // wave-uniform exit
    const int n0   = (wave - tm * nG) * (16 * NT);

    // A fragment: lanes 0-15 -> M=lane, K={k0,k0+1}; lanes 16-31 -> K={k0+2,k0+3}
    int rowA = (tm << 4) + (lane & 15);
    if (FLIP) {
        int b = rowA >> 11, l = rowA & (LSEQ - 1);
        rowA  = (b << 11) + (LSEQ - 1 - l);
    }
    const int kSub = (lane >> 4) << 1;          // 0 or 2
    const int nc   = lane & 15;

    const int xoff = rowA * lda + kSub;         // 32-bit element offsets
    int woff[NT];
#pragma unroll
    for (int i = 0; i < NT; ++i)
        woff[i] = (n0 + (i << 4) + nc) * K + kSub;

    v8f acc[NT];
#pragma unroll
    for (int i = 0; i < NT; ++i)
        acc[i] = v8f{0.f,0.f,0.f,0.f,0.f,0.f,0.f,0.f};

    // ---- software-pipelined K loop, prefetch distance 2 ----
    // buf0 holds even stages, buf1 holds odd stages (stage = k0/4).
    v2f a_buf0 = *(const v2f*)(X + xoff);
    v2f b_buf0[NT];
#pragma unroll
    for (int i = 0; i < NT; ++i)
        b_buf0[i] = *(const v2f*)(W + woff[i]);

    v2f a_buf1 = *(const v2f*)(X + xoff + 4);
    v2f b_buf1[NT];
#pragma unroll
    for (int i = 0; i < NT; ++i)
        b_buf1[i] = *(const v2f*)(W + woff[i] + 4);

#pragma unroll 2
    for (int k0 = 8; k0 <= K - 8; k0 += 8) {
        // --- half 1: load stage k0, consume stage k0-8 (buf0) ---
        v2f a_new0 = *(const v2f*)(X + xoff + k0);
        v2f b_new0[NT];
#pragma unroll
        for (int i = 0; i < NT; ++i)
            b_new0[i] = *(const v2f*)(W + woff[i] + k0);
#pragma unroll
        for (int i = 0; i < NT; ++i)
            acc[i] = __builtin_amdgcn_wmma_f32_16x16x4_f32(
                false, a_buf0, false, b_buf0[i], (short)0, acc[i], false, false);
        a_buf0 = a_new0;
#pragma unroll
        for (int i = 0; i < NT; ++i)
            b_buf0[i] = b_new0[i];

        // --- half 2: load stage k0+4, consume stage k0-4 (buf1) ---
        v2f a_new1 = *(const v2f*)(X + xoff + k0 + 4);
        v2f b_new1[NT];
#pragma unroll
        for (int i = 0; i < NT; ++i)
            b_new1[i] = *(const v2f*)(W + woff[i] + k0 + 4);
#pragma unroll
        for (int i = 0; i < NT; ++i)
            acc[i] = __builtin_amdgcn_wmma_f32_16x16x4_f32(
                false, a_buf1, false, b_buf1[i], (short)0, acc[i], false, false);
        a_buf1 = a_new1;
#pragma unroll
        for (int i = 0; i < NT; ++i)
            b_buf1[i] = b_new1[i];
    }

    // ---- drain: stages K-8 (buf0) and K-4 (buf1) ----
#pragma unroll
    for (int i = 0; i < NT; ++i)
        acc[i] = __builtin_amdgcn_wmma_f32_16x16x4_f32(
            false, a_buf0, false, b_buf0[i], (short)0, acc[i], false, false);
#pragma unroll
    for (int i = 0; i < NT; ++i)
        acc[i] = __builtin_amdgcn_wmma_f32_16x16x4_f32(
            false, a_buf1, false, b_buf1[i], (short)0, acc[i], false, false);

    // D layout: lanes 0-15: VGPR r -> M=r, N=lane ; lanes 16-31: M=r+8, N=lane-16
    const int rbase = (tm << 4) + ((lane >> 4) << 3);
#pragma unroll
    for (int i = 0; i < NT; ++i) {
        const int col = n0 + (i << 4) + nc;
        float bv = (EPI >= 1) ? bias[col] : 0.f;
#pragma unroll
        for (int r = 0; r < 8; ++r) {
            float v = acc[i][r];
            if (EPI >= 1) v += bv;
            if (EPI == 2) v = fmaxf(v, 0.f);
            if (EPI == 3) v = softplus_f(v);
            Y[(size_t)(rbase + r) * ldy + col] = v;
        }
    }
}

// ---------------------------------------------------------------------------
// Depthwise conv (k=2, causal) + SiLU.  xz rows: [xi(256) | z(256)].
// ---------------------------------------------------------------------------
__global__ __launch_bounds__(256) void conv_silu_kernel(
    const float* __restrict__ xz,
    const float* __restrict__ cw,   // (DINNER,2)
    const float* __restrict__ cb,
    float* __restrict__ xc)
{
    int idx = blockIdx.x * 256 + threadIdx.x;     // < NROWS*DINNER
    int d   = idx & 255;
    int row = idx >> 8;
    int t   = row & (LSEQ - 1);
    float xi = xz[(size_t)row * (2*DINNER) + d];
    float xp = (t > 0) ? xz[(size_t)(row - 1) * (2*DINNER) + d] : 0.f;
    float v  = xp * cw[d*2 + 0] + xi * cw[d*2 + 1] + cb[d];
    xc[idx]  = silu_f(v);
}

// ---------------------------------------------------------------------------
// Selective scan. One block per (b,d) channel; thread = one state index s.
// Fused skip (+u*D) and gate (*silu(z)).
// ---------------------------------------------------------------------------
__global__ __launch_bounds__(256) void scan_kernel(
    const float* __restrict__ dbc,    // (B*L, 528): [dt_lo(16)|B(256)|C(256)]
    const float* __restrict__ dtv,    // (B*L, 256) softplus'd dt
    const float* __restrict__ xc,     // (B*L, 256) u
    const float* __restrict__ xz,     // (B*L, 512) z in cols 256..511
    const float* __restrict__ Alog,   // (256,256)
    const float* __restrict__ Dp,     // (256,)
    float* __restrict__ y)            // (B*L, 256)
{
    const int d = blockIdx.x & 255;
    const int b = blockIdx.x >> 8;
    const int s = threadIdx.x;
    const int lane = threadIdx.x & 31;
    const int wid  = threadIdx.x >> 5;

    const float A_s = -__expf(Alog[d * DSTATE + s]);
    const float Dd  = Dp[d];
    float h = 0.f;

    __shared__ float red[8];

    for (int t = 0; t < LSEQ; ++t) {
        const int row = (b << 11) + t;
        const float dt_t = dtv[(size_t)row * DINNER + d];
        const float u_t  = xc [(size_t)row * DINNER + d];
        const float Bt   = dbc[(size_t)row * DXP + DTRANK + s];
        const float Ct   = dbc[(size_t)row * DXP + DTRANK + DSTATE + s];

        h = __expf(dt_t * A_s) * h + (dt_t * u_t) * Bt;
        float p = h * Ct;

#pragma unroll
        for (int off = 16; off > 0; off >>= 1)
            p += __shfl_down(p, off, 32);
        if (lane == 0) red[wid] = p;
        __syncthreads();
        if (threadIdx.x == 0) {
            float sum = red[0]+red[1]+red[2]+red[3]+red[4]+red[5]+red[6]+red[7];
            float zv  = xz[(size_t)row * (2*DINNER) + DINNER + d];
            y[(size_t)row * DINNER + d] = (sum + u_t * Dd) * (zv / (1.f + __expf(-zv)));
        }
        __syncthreads();
    }
}

// ---------------------------------------------------------------------------
// LayerNorm helpers
// ---------------------------------------------------------------------------
__device__ __forceinline__ float block_sum256(float v)
{
    __shared__ float sm[8];
    const int lane = threadIdx.x & 31, wid = threadIdx.x >> 5;
#pragma unroll
    for (int off = 16; off > 0; off >>= 1)
        v += __shfl_down(v, off, 32);
    if (lane == 0) sm[wid] = v;
    __syncthreads();
    if (threadIdx.x == 0) {
        float s = 0.f;
        for (int i = 0; i < 8; ++i) s += sm[i];
        sm[0] = s;
    }
    __syncthreads();
    float r = sm[0];
    __syncthreads();
    return r;
}

// out = LN(xa + xb [+ xcf flipped]) ; 1 block per row, 256 threads.
template<bool FLIPB>
__global__ __launch_bounds__(256) void add_ln_kernel(
    const float* __restrict__ xa,
    const float* __restrict__ xb,
    const float* __restrict__ xcf,
    const float* __restrict__ g,
    const float* __restrict__ be,
    float* __restrict__ out)
{
    const int row = blockIdx.x;
    const int c   = threadIdx.x;
    float v = xa[(size_t)row * DMODEL + c] + xb[(size_t)row * DMODEL + c];
    if (FLIPB) {
        int l = row & (LSEQ - 1), b = row >> 11;
        int rf = (b << 11) + (LSEQ - 1 - l);
        v += xcf[(size_t)rf * DMODEL + c];
    }
    float mean = block_sum256(v) * (1.f / DMODEL);
    float d    = v - mean;
    float var  = block_sum256(d * d) * (1.f / DMODEL);
    out[(size_t)row * DMODEL + c] = d * rsqrtf(var + 1e-5f) * g[c] + be[c];
}

// ---------------------------------------------------------------------------
// Launch
// ---------------------------------------------------------------------------
static inline int gemm_blocks(int M, int Ncols, int NT)
{
    int nG = Ncols / (16 * NT);
    int waves = (M / 16) * nG;
    return (waves + 7) / 8;
}

extern "C" void kernel_launch(void* const* d_in, const int* in_sizes, int n_in,
                              void* d_out, int out_size, void* d_ws, size_t ws_size,
                              hipStream_t stream) {
    (void)in_sizes; (void)n_in; (void)out_size; (void)ws_size;
    const float* x          = (const float*)d_in[0];
    const float* mf_in_w    = (const float*)d_in[1];
    const float* mf_conv_w  = (const float*)d_in[2];
    const float* mf_conv_b  = (const float*)d_in[3];
    const float* mf_xproj_w = (const float*)d_in[4];
    const float* mf_dt_w    = (const float*)d_in[5];
    const float* mf_dt_b    = (const float*)d_in[6];
    const float* mf_Alog    = (const float*)d_in[7];
    const float* mf_D       = (const float*)d_in[8];
    const float* mf_out_w   = (const float*)d_in[9];
    const float* mb_in_w    = (const float*)d_in[10];
    const float* mb_conv_w  = (const float*)d_in[11];
    const float* mb_conv_b  = (const float*)d_in[12];
    const float* mb_xproj_w = (const float*)d_in[13];
    const float* mb_dt_w    = (const float*)d_in[14];
    const float* mb_dt_b    = (const float*)d_in[15];
    const float* mb_Alog    = (const float*)d_in[16];
    const float* mb_D       = (const float*)d_in[17];
    const float* mb_out_w   = (const float*)d_in[18];
    const float* n1_g       = (const float*)d_in[19];
    const float* n1_b       = (const float*)d_in[20];
    const float* n2_g       = (const float*)d_in[21];
    const float* n2_b       = (const float*)d_in[22];
    const float* w1         = (const float*)d_in[23];
    const float* b1         = (const float*)d_in[24];
    const float* w2         = (const float*)d_in[25];
    const float* b2         = (const float*)d_in[26];

    float* ws = (float*)d_ws;
    // Workspace layout (floats); later buffers reuse dead earlier regions.
    float* xzf  = ws + 0;            // 4096*512
    float* xzb  = ws + 2097152;      // 4096*512
    float* xcf  = ws + 4194304;      // 4096*256
    float* xcb  = ws + 5242880;      // 4096*256
    float* dbcf = ws + 6291456;      // 4096*528
    float* dbcb = ws + 8454144;      // 4096*528
    float* dtf  = ws + 10616832;     // 4096*256
    float* dtb  = ws + 11665408;     // 4096*256
    float* yf   = ws + 12713984;     // 4096*256
    float* yb   = ws + 13762560;     // 4096*256
    float* xn1  = ws + 14811136;     // 4096*256
    float* of   = dbcf;              // reuse: dbc dead after scan
    float* ob   = dbcb;              // reuse
    float* hid  = xzf;               // reuse: xz dead after scan (4096*1024 fits xzf+xzb)
    float* fo   = dtf;               // reuse

    // in_proj (backward direction reads x time-flipped per batch)
    gemm_wmma_f32<0,false,4><<<gemm_blocks(NROWS, 2*DINNER, 4), 256, 0, stream>>>(
        x, DMODEL, mf_in_w, nullptr, xzf, 2*DINNER, NROWS, 2*DINNER, DMODEL);
    gemm_wmma_f32<0,true,4><<<gemm_blocks(NROWS, 2*DINNER, 4), 256, 0, stream>>>(
        x, DMODEL, mb_in_w, nullptr, xzb, 2*DINNER, NROWS, 2*DINNER, DMODEL);

    // causal depthwise conv + SiLU
    conv_silu_kernel<<<NROWS, 256, 0, stream>>>(xzf, mf_conv_w, mf_conv_b, xcf);
    conv_silu_kernel<<<NROWS, 256, 0, stream>>>(xzb, mb_conv_w, mb_conv_b, xcb);

    // x_proj -> [dt_lo | B | C]  (528 = 512 main + 16 tail columns)
    gemm_wmma_f32<0,false,4><<<gemm_blocks(NROWS, 512, 4), 256, 0, stream>>>(
        xcf, DINNER, mf_xproj_w, nullptr, dbcf, DXP, NROWS, 512, DINNER);
    gemm_wmma_f32<0,false,1><<<gemm_blocks(NROWS, 16, 1), 256, 0, stream>>>(
        xcf, DINNER, mf_xproj_w + (size_t)512 * DINNER, nullptr, dbcf + 512, DXP, NROWS, 16, DINNER);
    gemm_wmma_f32<0,false,4><<<gemm_blocks(NROWS, 512, 4), 256, 0, stream>>>(
        xcb, DINNER, mb_xproj_w, nullptr, dbcb, DXP, NROWS, 512, DINNER);
    gemm_wmma_f32<0,false,1><<<gemm_blocks(NROWS, 16, 1), 256, 0, stream>>>(
        xcb, DINNER, mb_xproj_w + (size_t)512 * DINNER, nullptr, dbcb + 512, DXP, NROWS, 16, DINNER);

    // dt = softplus(dt_lo @ dt_w.T + dt_b)   (X = first 16 cols of dbc, lda=528)
    gemm_wmma_f32<3,false,4><<<gemm_blocks(NROWS, DINNER, 4), 256, 0, stream>>>(
        dbcf, DXP, mf_dt_w, mf_dt_b, dtf, DINNER, NROWS, DINNER, DTRANK);
    gemm_wmma_f32<3,false,4><<<gemm_blocks(NROWS, DINNER, 4), 256, 0, stream>>>(
        dbcb, DXP, mb_dt_w, mb_dt_b, dtb, DINNER, NROWS, DINNER, DTRANK);

    // selective scan (fused +u*D and *silu(z))
    scan_kernel<<<2*DINNER, 256, 0, stream>>>(dbcf, dtf, xcf, xzf, mf_Alog, mf_D, yf);
    scan_kernel<<<2*DINNER, 256, 0, stream>>>(dbcb, dtb, xcb, xzb, mb_Alog, mb_D, yb);

    // out_proj
    gemm_wmma_f32<0,false,4><<<gemm_blocks(NROWS, DMODEL, 4), 256, 0, stream>>>(
        yf, DINNER, mf_out_w, nullptr, of, DMODEL, NROWS, DMODEL, DINNER);
    gemm_wmma_f32<0,false,4><<<gemm_blocks(NROWS, DMODEL, 4), 256, 0, stream>>>(
        yb, DINNER, mb_out_w, nullptr, ob, DMODEL, NROWS, DMODEL, DINNER);

    // residual + flip(bwd) + LayerNorm1
    add_ln_kernel<true><<<NROWS, 256, 0, stream>>>(x, of, ob, n1_g, n1_b, xn1);

    // FFN
    gemm_wmma_f32<2,false,4><<<gemm_blocks(NROWS, DFF, 4), 256, 0, stream>>>(
        xn1, DMODEL, w1, b1, hid, DFF, NROWS, DFF, DMODEL);
    gemm_wmma_f32<1,false,4><<<gemm_blocks(NROWS, DMODEL, 4), 256, 0, stream>>>(
        hid, DFF, w2, b2, fo, DMODEL, NROWS, DMODEL, DFF);

    // residual + LayerNorm2 -> output
    add_ln_kernel<false><<<NROWS, 256, 0, stream>>>(
        xn1, fo, nullptr, n2_g, n2_b, (float*)d_out);
}